// FocalLoss_rankingloss_prob_label_41755672052616
// MI455X (gfx1250) — compile-verified
//
#include <hip/hip_runtime.h>
#include <hip/hip_bf16.h>
#include <cstdint>
#include <cstddef>

// FocalLoss + ranking loss, B x C = 2097152 x 7, f32.
// Memory-bound (~90 MB @ 23.3 TB/s => ~3.9 us floor). No matmul structure =>
// WMMA inapplicable; the CDNA5 path exercised here is the async global->LDS
// data mover (ASYNCcnt) for staging 28-byte rows, wave32 per-wave staging
// with no workgroup barrier.

#define NTHREADS 256
#define CDIM 7

// Low 32 bits of a flat pointer to __shared__ == LDS byte offset
__device__ __forceinline__ unsigned lds_off(const void* p) {
  return (unsigned)(uintptr_t)p;
}

// 16B-per-lane async copy global -> LDS (honors EXEC; tracked by ASYNCcnt).
__device__ __forceinline__ void async_copy16(unsigned lds_dst, const void* gsrc) {
  asm volatile("global_load_async_to_lds_b128 %0, %1, off"
               :: "v"(lds_dst), "v"(gsrc)
               : "memory");
}

__device__ __forceinline__ void wait_async0() {
#if __has_builtin(__builtin_amdgcn_s_wait_asynccnt)
  __builtin_amdgcn_s_wait_asynccnt(0);
  asm volatile("" ::: "memory");
#else
  asm volatile("s_wait_asynccnt 0" ::: "memory");
#endif
}

// Core per-row math. lab in [0,1]; x[] are the 7 logits; m[] the mask bits.
__device__ __forceinline__ float row_loss(const float* x, const unsigned* m, float lab) {
  float p[CDIM], f[CDIM], fz[CDIM];
  float minp = __builtin_inff();
#pragma unroll
  for (int c = 0; c < CDIM; ++c) {
    float xc = x[c];
    // softplus(x) = max(x,0) + log1p(exp(-|x|)); e in (0,1] so log(1+e) is safe.
    float e    = __expf(-__builtin_fabsf(xc));
    float sp   = fmaxf(xc, 0.f) + __logf(1.f + e);
    float emsp = __expf(-sp);          // = 1 - sigmoid(x)
    float prob = 1.f - emsp;           // sigmoid(x)
    float bce  = sp - xc * lab;        // >= 0 always
    float t    = 1.f - __expf(-bce);
    f[c]  = t * t * bce;               // focal (gamma = 2)
    float bz = fminf(sp, 100.f);       // bce_zero = clip(softplus(x),0,100)
    float tz = (sp <= 100.f) ? prob : 1.f;  // 1 - exp(-100) == 1.0f in fp32
    fz[c] = tz * tz * bz;              // focal_zero
    p[c]  = prob;
    if (m[c] && prob < minp) minp = prob;
  }
  float loss = 0.f;
#pragma unroll
  for (int c = 0; c < CDIM; ++c) {
    // masked: focal; unmasked: focal_zero only where prob > masked-min (sign>0)
    loss += m[c] ? f[c] : ((p[c] > minp) ? fz[c] : 0.f);
  }
  return loss;
}

__global__ __launch_bounds__(NTHREADS) void focal_rank_kernel(
    const float* __restrict__ logit,
    const float* __restrict__ label,
    const unsigned char* __restrict__ mask,
    float* __restrict__ out,
    int nRows)
{
  __shared__ float         shLogit[NTHREADS * CDIM];  // 7168 B
  __shared__ unsigned char shMask [NTHREADS * CDIM];  // 1792 B

  const int tid  = threadIdx.x;
  const int wave = tid >> 5;
  const int lane = tid & 31;
  const int rowBase = blockIdx.x * NTHREADS + wave * 32;  // 32 rows per wave
  const int row     = rowBase + lane;

  if (rowBase + 32 <= nRows) {
    // ---- fast path: wave-private async staging, no block barrier ----
    const char* gLog = (const char*)logit + (size_t)rowBase * (CDIM * 4); // 896 B, 16B-aligned
    const char* gMsk = (const char*)mask  + (size_t)rowBase * CDIM;       // 224 B, 16B-aligned
    const unsigned lLog = lds_off(&shLogit[wave * 32 * CDIM]);
    const unsigned lMsk = lds_off(&shMask [wave * 32 * CDIM]);

    async_copy16(lLog + lane * 16, gLog + lane * 16);                          // 512 B
    if (lane < 24) async_copy16(lLog + 512 + lane * 16, gLog + 512 + lane * 16); // 384 B
    if (lane < 14) async_copy16(lMsk + lane * 16, gMsk + lane * 16);             // 224 B

    const float lab = label[row];   // overlaps the async copies
    wait_async0();

    const float*         rL = &shLogit[(size_t)(wave * 32 + lane) * CDIM];
    const unsigned char* rM = &shMask [(size_t)(wave * 32 + lane) * CDIM];
    float    x[CDIM];
    unsigned m[CDIM];
#pragma unroll
    for (int c = 0; c < CDIM; ++c) { x[c] = rL[c]; m[c] = rM[c]; }

    out[row] = row_loss(x, m, lab);
  } else if (row < nRows) {
    // ---- tail fallback (unused for B = 2^21, kept for generality) ----
    const float*         gL = logit + (size_t)row * CDIM;
    const unsigned char* gM = mask  + (size_t)row * CDIM;
    float    x[CDIM];
    unsigned m[CDIM];
#pragma unroll
    for (int c = 0; c < CDIM; ++c) { x[c] = gL[c]; m[c] = gM[c]; }
    out[row] = row_loss(x, m, label[row]);
  }
}

extern "C" void kernel_launch(void* const* d_in, const int* in_sizes, int n_in,
                              void* d_out, int out_size, void* d_ws, size_t ws_size,
                              hipStream_t stream) {
  (void)n_in; (void)d_ws; (void)ws_size; (void)out_size;
  const float*         logit = (const float*)d_in[0];
  const float*         label = (const float*)d_in[1];
  const unsigned char* mask  = (const unsigned char*)d_in[2];  // jnp.bool_ = 1 byte/elt
  float*               out   = (float*)d_out;

  const int nRows  = in_sizes[1];                 // label has B elements
  const int blocks = (nRows + NTHREADS - 1) / NTHREADS;
  hipLaunchKernelGGL(focal_rank_kernel, dim3(blocks), dim3(NTHREADS), 0, stream,
                     logit, label, mask, out, nRows);
}